// LTU_5918464934238
// MI455X (gfx1250) — compile-verified
//
#include <hip/hip_runtime.h>

typedef __attribute__((ext_vector_type(2))) float v2f;
typedef __attribute__((ext_vector_type(8))) float v8f;

#define N_INP_C 4096   // columns of W / length of x
// rows handled per block: 4 waves * 16 rows = 64

__global__ __launch_bounds__(128) void ltu_wmma_kernel(
    const float* __restrict__ x, const float* __restrict__ W,
    float* __restrict__ out) {
  __shared__ float xs[N_INP_C];

  const int tid = threadIdx.x;

  // ---- Stage x (16 KB) into LDS via the CDNA5 async-to-LDS path ----------
  // 128 threads x 16 B = 2 KB per sweep; 8 sweeps via the shared 24-bit
  // immediate offset (ISA adds INST_OFFSET to BOTH the LDS and global addr).
  {
    unsigned l = (unsigned)(unsigned long long)(uintptr_t)xs + (unsigned)tid * 16u;
    unsigned long long g =
        (unsigned long long)(uintptr_t)x + (unsigned long long)tid * 16ull;
#define LTU_ASYNC_CP(OFF)                                                  \
    asm volatile("global_load_async_to_lds_b128 %0, %1, off offset:" #OFF  \
                 :: "v"(l), "v"(g) : "memory")
    LTU_ASYNC_CP(0);
    LTU_ASYNC_CP(2048);
    LTU_ASYNC_CP(4096);
    LTU_ASYNC_CP(6144);
    LTU_ASYNC_CP(8192);
    LTU_ASYNC_CP(10240);
    LTU_ASYNC_CP(12288);
    LTU_ASYNC_CP(14336);
#undef LTU_ASYNC_CP
    asm volatile("s_wait_asynccnt 0" ::: "memory");
  }
  __syncthreads();

  const int lane = tid & 31;
  const int wave = tid >> 5;
  const int rowBase = (blockIdx.x * 4 + wave) * 16;
  const int r  = lane & 15;         // row within the 16-row tile (A-matrix M)
  const int hk = (lane >> 4) << 1;  // K sub-offset: 0 (lanes 0-15) or 2 (lanes 16-31)

  const float* wrow = W + (size_t)(rowBase + r) * N_INP_C + hk;

  v8f acc0 = {}, acc1 = {}, acc2 = {}, acc3 = {};
  unsigned neg = 0;

  for (int kb = 0; kb < N_INP_C; kb += 16) {
    __builtin_prefetch((const void*)(wrow + kb + 2048), 0, 1);

    // A fragments: per-lane contiguous 8B of this lane's row
    v2f a0 = *reinterpret_cast<const v2f*>(wrow + kb + 0);
    v2f a1 = *reinterpret_cast<const v2f*>(wrow + kb + 4);
    v2f a2 = *reinterpret_cast<const v2f*>(wrow + kb + 8);
    v2f a3 = *reinterpret_cast<const v2f*>(wrow + kb + 12);

    // B fragments: x[k] broadcast into all 16 columns
    v2f b0 = { xs[kb + 0  + hk], xs[kb + 1  + hk] };
    v2f b1 = { xs[kb + 4  + hk], xs[kb + 5  + hk] };
    v2f b2 = { xs[kb + 8  + hk], xs[kb + 9  + hk] };
    v2f b3 = { xs[kb + 12 + hk], xs[kb + 13 + hk] };

    // negative-weight count via sign bits (2 VALU/float, fuses into v_add3)
    neg += (__float_as_uint(a0.x) >> 31) + (__float_as_uint(a0.y) >> 31)
         + (__float_as_uint(a1.x) >> 31) + (__float_as_uint(a1.y) >> 31)
         + (__float_as_uint(a2.x) >> 31) + (__float_as_uint(a2.y) >> 31)
         + (__float_as_uint(a3.x) >> 31) + (__float_as_uint(a3.y) >> 31);

    // D = A*B + C ; 4 independent accumulators break the RAW chain
    acc0 = __builtin_amdgcn_wmma_f32_16x16x4_f32(false, a0, false, b0,
                                                 (short)0, acc0, false, false);
    acc1 = __builtin_amdgcn_wmma_f32_16x16x4_f32(false, a1, false, b1,
                                                 (short)0, acc1, false, false);
    acc2 = __builtin_amdgcn_wmma_f32_16x16x4_f32(false, a2, false, b2,
                                                 (short)0, acc2, false, false);
    acc3 = __builtin_amdgcn_wmma_f32_16x16x4_f32(false, a3, false, b3,
                                                 (short)0, acc3, false, false);
  }

  v8f acc = (acc0 + acc1) + (acc2 + acc3);

  // Fold K-halves of the negative count: row r total = lane r + lane r+16
  int negc = (int)neg;
  negc += __shfl_xor(negc, 16, 32);

  // D layout: lanes 0-15 hold rows 0-7 in v[0..7]; lanes 16-31 hold rows 8-15.
  // All 16 columns are identical (B columns identical), so lanes 0 and 16
  // carry valid scores and do the write-out.
  const int base = (lane >> 4) << 3;  // 0 or 8
  float res[8];
#pragma unroll
  for (int j = 0; j < 8; ++j) {
    int cj = __shfl(negc, base + j, 32);           // count for row base+j
    float tau = 0.6f * (float)N_INP_C - (float)cj; // matches f32 reference
    res[j] = (acc[j] < tau) ? 0.0f : 1.0f;
  }
  if (lane == 0 || lane == 16) {
    float* o = out + rowBase + base;
#pragma unroll
    for (int j = 0; j < 8; ++j) o[j] = res[j];
  }
}

extern "C" void kernel_launch(void* const* d_in, const int* in_sizes, int n_in,
                              void* d_out, int out_size, void* d_ws, size_t ws_size,
                              hipStream_t stream) {
  const float* x = (const float*)d_in[0];   // input  [4096]
  const float* W = (const float*)d_in[1];   // weight [8192, 4096]
  float* out = (float*)d_out;               // [8192]

  const int n_rows = out_size;              // 8192
  dim3 block(128);                          // 4 waves
  dim3 grid(n_rows / 64);                   // 64 rows per block -> 128 blocks
  ltu_wmma_kernel<<<grid, block, 0, stream>>>(x, W, out);
}